// BipartiteGNN_16947940950559
// MI455X (gfx1250) — compile-verified
//
#include <hip/hip_runtime.h>

typedef __attribute__((ext_vector_type(2))) float v2f;
typedef __attribute__((ext_vector_type(8))) float v8f;

#define BM 128
#define BM2 64
#define BN 128
#define BK 32
#define AS_STRIDE 36    // 36*4  = 144 B row stride (16B aligned)
#define BS_STRIDE 132   // 132*4 = 528 B row stride (16B aligned)

// -------------------------------------------------------------------------
// Y[M,N] = act(X[M,K] @ W[K,N] + bias[N]);  K % 32 == 0, N % 128 == 0.
// 256 threads = 8 waves arranged 4(M) x 2(N); each wave computes 32x64 via
// 2x4 tiles of V_WMMA_F32_16X16X4_F32.
// -------------------------------------------------------------------------
__global__ __launch_bounds__(256)
void gnn_gemm_bias_act(const float* __restrict__ X, const float* __restrict__ W,
                       const float* __restrict__ bias, float* __restrict__ Y,
                       int M, int K, int N, int doRelu) {
  __shared__ float As[BM * AS_STRIDE];
  __shared__ float Bs[BK * BS_STRIDE];

  const int tid   = threadIdx.x;
  const int lane  = tid & 31;
  const int wave  = tid >> 5;
  const int waveM = wave & 3;         // 0..3 -> 32-row strip
  const int waveN = wave >> 2;        // 0..1 -> 64-col strip
  const int lr    = lane & 15;
  const int kHalf = (lane >> 4) << 1; // lanes 0-15 -> K+{0,1}; 16-31 -> K+{2,3}

  const int m0 = blockIdx.x * BM;
  const int n0 = blockIdx.y * BN;

  v8f acc[2][4];
#pragma unroll
  for (int mt = 0; mt < 2; ++mt)
#pragma unroll
    for (int nt = 0; nt < 4; ++nt)
      acc[mt][nt] = (v8f)0.0f;

  for (int k0 = 0; k0 < K; k0 += BK) {
#pragma unroll
    for (int i = 0; i < 4; ++i) {          // A tile: 128 x 32
      int s  = tid + i * 256;
      int r  = s >> 3;
      int c4 = (s & 7) << 2;
      float4 v = make_float4(0.f, 0.f, 0.f, 0.f);
      int gr = m0 + r;
      if (gr < M)
        v = *(const float4*)(X + (size_t)gr * K + (k0 + c4));
      *(float4*)(As + r * AS_STRIDE + c4) = v;
    }
#pragma unroll
    for (int i = 0; i < 4; ++i) {          // B tile: 32 x 128
      int s  = tid + i * 256;
      int r  = s >> 5;
      int c4 = (s & 31) << 2;
      float4 v = *(const float4*)(W + (size_t)(k0 + r) * N + (n0 + c4));
      *(float4*)(Bs + r * BS_STRIDE + c4) = v;
    }
    __syncthreads();

#pragma unroll
    for (int kk = 0; kk < BK; kk += 4) {
      v2f a[2];
      v2f b[4];
#pragma unroll
      for (int mt = 0; mt < 2; ++mt) {
        int m = waveM * 32 + mt * 16 + lr;
        const float* p = As + m * AS_STRIDE + kk + kHalf;
        a[mt][0] = p[0];
        a[mt][1] = p[1];
      }
#pragma unroll
      for (int nt = 0; nt < 4; ++nt) {
        int n = waveN * 64 + nt * 16 + lr;
        const float* p = Bs + (kk + kHalf) * BS_STRIDE + n;
        b[nt][0] = p[0];
        b[nt][1] = p[BS_STRIDE];
      }
#pragma unroll
      for (int mt = 0; mt < 2; ++mt)
#pragma unroll
        for (int nt = 0; nt < 4; ++nt)
          acc[mt][nt] = __builtin_amdgcn_wmma_f32_16x16x4_f32(
              false, a[mt], false, b[nt], (short)0, acc[mt][nt], false, false);
    }
    __syncthreads();
  }

  const int rowHalf = (lane >> 4) << 3;
#pragma unroll
  for (int mt = 0; mt < 2; ++mt) {
#pragma unroll
    for (int nt = 0; nt < 4; ++nt) {
      int col  = n0 + waveN * 64 + nt * 16 + lr;
      float bv = bias[col];
#pragma unroll
      for (int v = 0; v < 8; ++v) {
        int row = m0 + waveM * 32 + mt * 16 + rowHalf + v;
        float val = acc[mt][nt][v] + bv;
        if (doRelu) val = fmaxf(val, 0.0f);
        if (row < M) Y[(size_t)row * N + col] = val;
      }
    }
  }
}

// -------------------------------------------------------------------------
// Dual-output layer-1 kernel: one pass over X produces
//   Y1 = relu(X @ W1 + b1)  and  Y2 = relu(X @ W2 + b2).
// Block tile: 64 rows (shared A in LDS) x 128 cols; 8 waves split 4+4 over
// the two outputs (waves w<4 -> W1/Y1, w>=4 -> W2/Y2), each wave 32x64.
// Halves the HBM reads of the big feature matrices.
// -------------------------------------------------------------------------
__global__ __launch_bounds__(256)
void gnn_gemm_dual(const float* __restrict__ X,
                   const float* __restrict__ W1, const float* __restrict__ b1,
                   float* __restrict__ Y1,
                   const float* __restrict__ W2, const float* __restrict__ b2,
                   float* __restrict__ Y2,
                   int M, int K, int N) {
  __shared__ float As[BM2 * AS_STRIDE];        // 64 x 32
  __shared__ float Bs[2][BK * BS_STRIDE];      // two 32 x 128 tiles

  const int tid     = threadIdx.x;
  const int lane    = tid & 31;
  const int wave    = tid >> 5;
  const int waveOut = wave >> 2;      // 0 -> (W1,Y1), 1 -> (W2,Y2)
  const int wq      = wave & 3;
  const int waveM   = wq & 1;         // 0..1 -> 32-row strip
  const int waveN   = wq >> 1;        // 0..1 -> 64-col strip
  const int lr      = lane & 15;
  const int kHalf   = (lane >> 4) << 1;

  const int m0 = blockIdx.x * BM2;
  const int n0 = blockIdx.y * BN;

  v8f acc[2][4];
#pragma unroll
  for (int mt = 0; mt < 2; ++mt)
#pragma unroll
    for (int nt = 0; nt < 4; ++nt)
      acc[mt][nt] = (v8f)0.0f;

  const float* Bw = &Bs[waveOut][0];

  for (int k0 = 0; k0 < K; k0 += BK) {
#pragma unroll
    for (int i = 0; i < 2; ++i) {            // A tile: 64 x 32
      int s  = tid + i * 256;
      int r  = s >> 3;
      int c4 = (s & 7) << 2;
      float4 v = make_float4(0.f, 0.f, 0.f, 0.f);
      int gr = m0 + r;
      if (gr < M)
        v = *(const float4*)(X + (size_t)gr * K + (k0 + c4));
      *(float4*)(As + r * AS_STRIDE + c4) = v;
    }
#pragma unroll
    for (int i = 0; i < 4; ++i) {            // B1 tile: 32 x 128
      int s  = tid + i * 256;
      int r  = s >> 5;
      int c4 = (s & 31) << 2;
      float4 v = *(const float4*)(W1 + (size_t)(k0 + r) * N + (n0 + c4));
      *(float4*)(&Bs[0][0] + r * BS_STRIDE + c4) = v;
    }
#pragma unroll
    for (int i = 0; i < 4; ++i) {            // B2 tile: 32 x 128
      int s  = tid + i * 256;
      int r  = s >> 5;
      int c4 = (s & 31) << 2;
      float4 v = *(const float4*)(W2 + (size_t)(k0 + r) * N + (n0 + c4));
      *(float4*)(&Bs[1][0] + r * BS_STRIDE + c4) = v;
    }
    __syncthreads();

#pragma unroll
    for (int kk = 0; kk < BK; kk += 4) {
      v2f a[2];
      v2f b[4];
#pragma unroll
      for (int mt = 0; mt < 2; ++mt) {
        int m = waveM * 32 + mt * 16 + lr;
        const float* p = As + m * AS_STRIDE + kk + kHalf;
        a[mt][0] = p[0];
        a[mt][1] = p[1];
      }
#pragma unroll
      for (int nt = 0; nt < 4; ++nt) {
        int n = waveN * 64 + nt * 16 + lr;
        const float* p = Bw + (kk + kHalf) * BS_STRIDE + n;
        b[nt][0] = p[0];
        b[nt][1] = p[BS_STRIDE];
      }
#pragma unroll
      for (int mt = 0; mt < 2; ++mt)
#pragma unroll
        for (int nt = 0; nt < 4; ++nt)
          acc[mt][nt] = __builtin_amdgcn_wmma_f32_16x16x4_f32(
              false, a[mt], false, b[nt], (short)0, acc[mt][nt], false, false);
    }
    __syncthreads();
  }

  const float* bias = waveOut ? b2 : b1;
  float*       Y    = waveOut ? Y2 : Y1;
  const int rowHalf = (lane >> 4) << 3;
#pragma unroll
  for (int mt = 0; mt < 2; ++mt) {
#pragma unroll
    for (int nt = 0; nt < 4; ++nt) {
      int col  = n0 + waveN * 64 + nt * 16 + lr;
      float bv = bias[col];
#pragma unroll
      for (int v = 0; v < 8; ++v) {
        int row = m0 + waveM * 32 + mt * 16 + rowHalf + v;
        float val = fmaxf(acc[mt][nt][v] + bv, 0.0f);   // layer-1 is always ReLU
        if (row < M) Y[(size_t)row * N + col] = val;
      }
    }
  }
}

// -------------------------------------------------------------------------
// Per edge: agg[recv[e]][c] += msg[send[e]][c], c = 0..255.
// One block per edge; 256 lanes carry the 256 channels. Hardware f32 atomic
// (global_atomic_add_f32, no-return, device scope -> resolves at L2 atomic
// units; message tables are L2-resident).
// -------------------------------------------------------------------------
__global__ __launch_bounds__(256)
void gnn_scatter_add(const int* __restrict__ recvIdx, const int* __restrict__ sendIdx,
                     const float* __restrict__ msg, float* __restrict__ agg, int nEdges) {
  int e = blockIdx.x;
  if (e >= nEdges) return;
  int r = recvIdx[e];
  int s = sendIdx[e];
  int c = threadIdx.x;
  float v = msg[(size_t)s * 256 + c];
  unsigned int off = ((unsigned int)r * 256u + (unsigned int)c) * 4u;
  asm volatile("global_atomic_add_f32 %0, %1, %2 scope:SCOPE_DEV"
               :
               : "v"(off), "v"(v), "s"(agg)
               : "memory");
}

extern "C" void kernel_launch(void* const* d_in, const int* in_sizes, int n_in,
                              void* d_out, int out_size, void* d_ws, size_t ws_size,
                              hipStream_t stream) {
  (void)in_sizes; (void)n_in; (void)out_size; (void)ws_size;

  constexpr int U = 50000, D = 10000, E = 200000;
  constexpr int DIN = 1024, HID = 256, OUT = 128;

  const float* user_feat = (const float*)d_in[0];
  const float* dish_feat = (const float*)d_in[1];
  const int*   edge_user = (const int*)d_in[2];
  const int*   edge_dish = (const int*)d_in[3];
  const float* W_ui = (const float*)d_in[4];   const float* b_ui = (const float*)d_in[5];
  const float* W_di = (const float*)d_in[6];   const float* b_di = (const float*)d_in[7];
  const float* W_um = (const float*)d_in[8];   const float* b_um = (const float*)d_in[9];
  const float* W_dm = (const float*)d_in[10];  const float* b_dm = (const float*)d_in[11];
  const float* W_uu = (const float*)d_in[12];  const float* b_uu = (const float*)d_in[13];
  const float* W_du = (const float*)d_in[14];  const float* b_du = (const float*)d_in[15];
  const float* W_up = (const float*)d_in[16];  const float* b_up = (const float*)d_in[17];
  const float* W_dp = (const float*)d_in[18];  const float* b_dp = (const float*)d_in[19];

  float* out_user = (float*)d_out;                 // [U, OUT]
  float* out_dish = out_user + (size_t)U * OUT;    // [D, OUT]

  // Workspace layout (floats): ~123 MB total
  float* ws      = (float*)d_ws;
  float* userAgg = ws;                              // U*HID  (init + scattered msgs)
  float* dishAgg = userAgg + (size_t)U * HID;       // D*HID
  float* msgUser = dishAgg + (size_t)D * HID;       // U*HID  (relu(user@W_dm+b))
  float* msgDish = msgUser + (size_t)U * HID;       // D*HID  (relu(dish@W_um+b))
  float* userUpd = msgUser;                         // reuse after scatters done
  float* dishUpd = msgDish;                         // reuse after scatters done

  dim3 blk(256);
  const int gU  = (U + BM - 1) / BM;    // 391
  const int gD  = (D + BM - 1) / BM;    // 79
  const int gU2 = (U + BM2 - 1) / BM2;  // 782
  const int gD2 = (D + BM2 - 1) / BM2;  // 157

  // ---- layer 1: one pass per feature matrix, two projections each ----
  gnn_gemm_dual<<<dim3(gU2, HID / BN), blk, 0, stream>>>(
      user_feat, W_ui, b_ui, userAgg, W_dm, b_dm, msgUser, U, DIN, HID);
  gnn_gemm_dual<<<dim3(gD2, HID / BN), blk, 0, stream>>>(
      dish_feat, W_di, b_di, dishAgg, W_um, b_um, msgDish, D, DIN, HID);

  // ---- edge scatter: agg += gathered per-node messages ----
  gnn_scatter_add<<<dim3(E), blk, 0, stream>>>(edge_user, edge_dish, msgDish, userAgg, E);
  gnn_scatter_add<<<dim3(E), blk, 0, stream>>>(edge_dish, edge_user, msgUser, dishAgg, E);

  // ---- layer 2: update + output projection ----
  gnn_gemm_bias_act<<<dim3(gU, HID / BN), blk, 0, stream>>>(userAgg, W_uu, b_uu, userUpd, U, HID, HID, 1);
  gnn_gemm_bias_act<<<dim3(gU, OUT / BN), blk, 0, stream>>>(userUpd, W_up, b_up, out_user, U, HID, OUT, 0);
  gnn_gemm_bias_act<<<dim3(gD, HID / BN), blk, 0, stream>>>(dishAgg, W_du, b_du, dishUpd, D, HID, HID, 1);
  gnn_gemm_bias_act<<<dim3(gD, OUT / BN), blk, 0, stream>>>(dishUpd, W_dp, b_dp, out_dish, D, HID, OUT, 0);
}